// myVMLSTM_Group_48172353192383
// MI455X (gfx1250) — compile-verified
//
#include <hip/hip_runtime.h>
#include <math.h>

// Problem sizes (fixed by the reference)
#define T_STEPS 512
#define NB      40      // batch
#define NI      1024    // input dim
#define NH      1024    // hidden dim
#define NWR     64      // input low rank
#define NUR     32      // recurrent low rank (x2 perms -> K=64)
#define M_ROWS  (T_STEPS * NB)   // 20480
#define TMPSZ   (NB * 2 * 64)    // 5120 floats per tmp buffer

typedef float v2f __attribute__((ext_vector_type(2)));
typedef float v8f __attribute__((ext_vector_type(8)));

// Native fp32 WMMA: D(16x16) = A(16x4) * B(4x16) + C, full fp32 precision.
__device__ __forceinline__ v8f wmma4(v2f a, v2f b, v8f c) {
    return __builtin_amdgcn_wmma_f32_16x16x4_f32(
        /*neg_a=*/false, a, /*neg_b=*/false, b,
        /*c_mod=*/(short)0, c, /*reuse_a=*/false, /*reuse_b=*/false);
}

__device__ __forceinline__ float sigmf(float x) { return 1.0f / (1.0f + __expf(-x)); }

// ---------------------------------------------------------------------------
// Init: copy h0,c0 into state buffers; zero tmp rotation buffers 1 and 2.
__global__ void vml_init(const float* __restrict__ h0, const float* __restrict__ c0,
                         float* __restrict__ hbuf, float* __restrict__ cbuf,
                         float* __restrict__ tmpb) {
    int i = blockIdx.x * blockDim.x + threadIdx.x;
    if (i < NB * NH) { hbuf[i] = h0[i]; cbuf[i] = c0[i]; }
    if (i < 2 * TMPSZ) tmpb[TMPSZ + i] = 0.0f;   // buffers 1 and 2
}

// ---------------------------------------------------------------------------
// dvec_x[j] = sum_r U_x[j%1024, r] * W_x[j, r]
// dvec_h[j] = sum_r U_h[0][flat h=j%1024][r] * V_h[0, j/2048, r, j%2048]
__global__ void vml_dvec(const float* __restrict__ Ux, const float* __restrict__ Wx,
                         const float* __restrict__ Uh, const float* __restrict__ Vh,
                         float* __restrict__ dvx, float* __restrict__ dvh) {
    int j = blockIdx.x * blockDim.x + threadIdx.x;
    if (j >= 4 * NH) return;
    int h = j & (NH - 1);
    float sx = 0.f;
    #pragma unroll 8
    for (int r = 0; r < NWR; ++r) sx += Ux[h * NWR + r] * Wx[j * NWR + r];
    dvx[j] = sx;
    int g = j >> 11;            // which group block of 2048
    int k = j & 2047;
    float sh = 0.f;
    #pragma unroll 8
    for (int r = 0; r < NUR; ++r)
        sh += Uh[h * NUR + r] * Vh[(size_t)(g * NUR + r) * 2048 + k];
    dvh[j] = sh;
}

// ---------------------------------------------------------------------------
// P[20480 x 64] = x[20480 x 1024] @ U_x[1024 x 64]   (fp32 WMMA, K-loop of 4)
// One wave computes a 16-row stripe across all 64 output columns (A reuse x4).
__global__ void vml_xu_gemm(const float* __restrict__ x, const float* __restrict__ Ux,
                            float* __restrict__ P) {
    const int wave = (blockIdx.x * blockDim.x + threadIdx.x) >> 5;
    const int lane = threadIdx.x & 31;
    const int hf   = lane >> 4;      // lane half: selects K pair {0,1} vs {2,3}
    const int lm   = lane & 15;
    const int m0   = wave * 16;
    if (m0 >= M_ROWS) return;

    v8f acc0 = {0,0,0,0,0,0,0,0}, acc1 = acc0, acc2 = acc0, acc3 = acc0;
    const float* xrow = x + (size_t)(m0 + lm) * NI;

    #pragma unroll 4
    for (int k = 0; k < NI; k += 4) {
        const int kk = k + 2 * hf;
        v2f a = *(const v2f*)(xrow + kk);                 // A(m, kk), A(m, kk+1)
        v2f b0, b1, b2, b3;
        b0.x = Ux[kk * NWR +  0 + lm]; b0.y = Ux[(kk + 1) * NWR +  0 + lm];
        b1.x = Ux[kk * NWR + 16 + lm]; b1.y = Ux[(kk + 1) * NWR + 16 + lm];
        b2.x = Ux[kk * NWR + 32 + lm]; b2.y = Ux[(kk + 1) * NWR + 32 + lm];
        b3.x = Ux[kk * NWR + 48 + lm]; b3.y = Ux[(kk + 1) * NWR + 48 + lm];
        acc0 = wmma4(a, b0, acc0);
        acc1 = wmma4(a, b1, acc1);
        acc2 = wmma4(a, b2, acc2);
        acc3 = wmma4(a, b3, acc3);
    }
    #pragma unroll
    for (int v = 0; v < 8; ++v) {
        const int row = m0 + v + 8 * hf;
        P[(size_t)row * NWR +  0 + lm] = acc0[v];
        P[(size_t)row * NWR + 16 + lm] = acc1[v];
        P[(size_t)row * NWR + 32 + lm] = acc2[v];
        P[(size_t)row * NWR + 48 + lm] = acc3[v];
    }
}

// ---------------------------------------------------------------------------
// Seed (t=0 only): tmp[b][g][p*32+r] = sum_d h0[b, ((g+p)&1)*512 + d] * U_h[p,g,d,r]
// 24 WMMA tiles (2p x 2g x 3 mtiles x 2 ntiles), K=512.
__global__ void vml_tmp(const float* __restrict__ hbuf, const float* __restrict__ Uh,
                        float* __restrict__ tmpb) {
    const int wave = blockIdx.x * (blockDim.x >> 5) + (threadIdx.x >> 5);  // 0..23
    const int lane = threadIdx.x & 31;
    const int hf = lane >> 4, lm = lane & 15;

    const int mt = wave % 3;
    const int rest = wave / 3;
    const int g  = rest & 1;
    const int p  = (rest >> 1) & 1;
    const int nt = rest >> 2;              // 0..1
    const int gsrc = (g + p) & 1;

    const int brow = mt * 16 + lm;
    const int bc = brow < NB ? brow : NB - 1;
    const float* arow  = hbuf + (size_t)bc * NH + gsrc * 512;
    const float* Bbase = Uh + (size_t)(p * 2 + g) * 512 * NUR;

    v8f acc = {0,0,0,0,0,0,0,0};
    #pragma unroll 4
    for (int k = 0; k < 512; k += 4) {
        const int kk = k + 2 * hf;
        v2f a = *(const v2f*)(arow + kk);
        v2f b;
        b.x = Bbase[kk * NUR + nt * 16 + lm];
        b.y = Bbase[(kk + 1) * NUR + nt * 16 + lm];
        acc = wmma4(a, b, acc);
    }
    #pragma unroll
    for (int v = 0; v < 8; ++v) {
        const int bb = mt * 16 + v + 8 * hf;
        if (bb < NB)
            tmpb[(bb * 2 + g) * 64 + p * 32 + nt * 16 + lm] = acc[v];
    }
}

// ---------------------------------------------------------------------------
// Fused per-step kernel (the ONLY kernel on the 512-step critical path):
//  phase 0: zero its slice of tmp_zero (buffer for step t+2)
//  phase 1: gxw = P_t(40x64) @ W_x^T  and  lhw = tmp_cur(40x64) @ V_h   (8x16 WMMA)
//  phase 2: diag/bias terms, gates, c/h update; write h,c and out[t]
//  phase 3: stage h_new tile in LDS, 16 WMMAs (A from LDS) to form this tile's
//           K=16 partial of tmp(t+1); accumulate via global fp32 atomics.
__global__ void vml_step(int t,
                         const float* __restrict__ P,    const float* __restrict__ Wx,
                         const float* __restrict__ Vh,   const float* __restrict__ Uh,
                         const float* __restrict__ tmp_cur, float* __restrict__ tmp_next,
                         float* __restrict__ tmp_zero,
                         const float* __restrict__ x,
                         float* __restrict__ hbuf,       float* __restrict__ cbuf,
                         const float* __restrict__ dvx,  const float* __restrict__ dvh,
                         const float* __restrict__ bx,   const float* __restrict__ bh,
                         const float* __restrict__ diax, const float* __restrict__ diah,
                         float* __restrict__ out) {
    __shared__ float lsh[2 * 16 * 18];                 // per-wave 16x16 h tile, padded
    const int w    = (threadIdx.x >> 5) & 1;
    const int idx  = blockIdx.x * (blockDim.x >> 5) + (threadIdx.x >> 5);   // 0..191
    const int lane = threadIdx.x & 31;
    const int hf = lane >> 4, lm = lane & 15;
    const int ut = idx & 63;        // 64 u-tiles of 16
    const int mt = idx >> 6;        // 3 row tiles of 16
    const int u0 = ut * 16;

    // phase 0: zero the tmp buffer that step t+2 will read (27 floats per wave)
    {
        const int z = idx * 27 + lane;
        if (lane < 27 && z < TMPSZ) tmp_zero[z] = 0.0f;
    }

    const int brow = mt * 16 + lm;
    const int bc = brow < NB ? brow : NB - 1;

    const float* Pt  = P + (size_t)t * NB * NWR;
    const float* agx = Pt + (size_t)bc * NWR;                 // A rows for gx gemm

    // Prefetch next step's P rows (the only per-step stream not already L2-hot)
    if (t + 1 < T_STEPS) __builtin_prefetch(agx + NB * NWR, 0, 0);

    v8f sgx[4], sgh[4];
    #pragma unroll
    for (int gi = 0; gi < 4; ++gi) { sgx[gi] = (v8f){0,0,0,0,0,0,0,0}; sgh[gi] = sgx[gi]; }

    #pragma unroll
    for (int gi = 0; gi < 4; ++gi) {
        const int j0 = gi * NH + u0;
        // --- gx: lowered_x tile = P_t @ W_x^T ; B(c,n) = W_x[j0+n, c] (contiguous pairs)
        const float* Bw = Wx + (size_t)(j0 + lm) * NWR;
        #pragma unroll
        for (int k = 0; k < NWR; k += 4) {
            const int kk = k + 2 * hf;
            v2f a = *(const v2f*)(agx + kk);
            v2f b = *(const v2f*)(Bw + kk);
            sgx[gi] = wmma4(a, b, sgx[gi]);
        }
        // --- gh: lowered_h tile = tmp_cur @ V_h ; group g = gi>>1
        const int g    = gi >> 1;
        const int kcol = (gi & 1) * NH + u0 + lm;
        const float* agh = tmp_cur + (size_t)(bc * 2 + g) * 64;
        #pragma unroll
        for (int k = 0; k < 64; k += 4) {
            const int cc = k + 2 * hf;
            v2f a = *(const v2f*)(agh + cc);
            v2f b;
            b.x = Vh[(size_t)(((cc       >> 5) * 2 + g) * NUR + (cc       & 31)) * 2048 + kcol];
            b.y = Vh[(size_t)((((cc + 1) >> 5) * 2 + g) * NUR + ((cc + 1) & 31)) * 2048 + kcol];
            sgh[gi] = wmma4(a, b, sgh[gi]);
        }
    }

    // phase 2: elementwise epilogue; lane owns (b = mt*16 + v + 8*hf, u = u0+lm)
    const int u = u0 + lm;
    const float dxu = diax[u], dhu = diah[u];
    float dvxg[4], dvhg[4], bxg[4], bhg[4];
    #pragma unroll
    for (int gi = 0; gi < 4; ++gi) {
        const int j = gi * NH + u;
        dvxg[gi] = dvx[j]; dvhg[gi] = dvh[j]; bxg[gi] = bx[j]; bhg[gi] = bh[j];
    }

    #pragma unroll
    for (int v = 0; v < 8; ++v) {
        const int b = mt * 16 + v + 8 * hf;
        const int bcl = b < NB ? b : NB - 1;
        const float xv = x[((size_t)t * NB + bcl) * NI + u];
        const float hv = hbuf[(size_t)bcl * NH + u];
        const float cv = cbuf[(size_t)bcl * NH + u];
        float s[4];
        #pragma unroll
        for (int gi = 0; gi < 4; ++gi) {
            const float gx = sgx[gi][v] + dxu * xv - xv * dvxg[gi] + bxg[gi];
            const float gh = sgh[gi][v] + dhu * hv - hv * dvhg[gi] + bhg[gi];
            s[gi] = gx + gh;
        }
        const float ig = sigmf(s[0]);
        const float fg = sigmf(s[1]);
        const float og = sigmf(s[2]);
        const float ng = tanhf(s[3]);
        const float cn = fg * cv + ig * ng;
        const float hn = og * tanhf(cn);
        const bool ok = (b < NB);
        if (ok) {
            cbuf[(size_t)b * NH + u] = cn;
            hbuf[(size_t)b * NH + u] = hn;
            out[((size_t)t * NB + b) * NH + u] = hn;
        }
        lsh[w * 288 + (v + 8 * hf) * 18 + lm] = ok ? hn : 0.0f;  // zero invalid rows
    }
    __syncthreads();

    // phase 3: this tile's K=16 partial of tmp(t+1) = h_tile @ U_h slice, via WMMA.
    // u-slice maps to source half gsrc = u0>=512, offset d0 = u0&511. For each g,
    // the matching permutation is p = g ^ gsrc  (since gsrc == (g+p)&1).
    const int gsrc = (u0 >= 512) ? 1 : 0;
    const int d0   = u0 & 511;
    #pragma unroll
    for (int g = 0; g < 2; ++g) {
        const int p = g ^ gsrc;
        const float* Bbase = Uh + (size_t)((p * 2 + g) * 512 + d0) * NUR;
        v8f acc0 = {0,0,0,0,0,0,0,0}, acc1 = acc0;
        #pragma unroll
        for (int k = 0; k < 16; k += 4) {
            const int kk = k + 2 * hf;
            v2f a = *(const v2f*)&lsh[w * 288 + lm * 18 + kk];   // A(b_local, d_local)
            v2f b0, b1;
            b0.x = Bbase[kk * NUR + lm];      b0.y = Bbase[(kk + 1) * NUR + lm];
            b1.x = Bbase[kk * NUR + 16 + lm]; b1.y = Bbase[(kk + 1) * NUR + 16 + lm];
            acc0 = wmma4(a, b0, acc0);
            acc1 = wmma4(a, b1, acc1);
        }
        #pragma unroll
        for (int v = 0; v < 8; ++v) {
            const int b = mt * 16 + v + 8 * hf;
            if (b < NB) {
                float* dst = tmp_next + (b * 2 + g) * 64 + p * 32;
                unsafeAtomicAdd(dst + lm,      acc0[v]);
                unsafeAtomicAdd(dst + 16 + lm, acc1[v]);
            }
        }
    }
}

// ---------------------------------------------------------------------------
// Final: h_f = hbuf, c_f = cbuf appended after outputs.
__global__ void vml_fin(const float* __restrict__ hbuf, const float* __restrict__ cbuf,
                        float* __restrict__ out) {
    int i = blockIdx.x * blockDim.x + threadIdx.x;
    if (i < NB * NH) {
        out[(size_t)T_STEPS * NB * NH + i] = hbuf[i];
        out[(size_t)T_STEPS * NB * NH + NB * NH + i] = cbuf[i];
    }
}

// ---------------------------------------------------------------------------
extern "C" void kernel_launch(void* const* d_in, const int* in_sizes, int n_in,
                              void* d_out, int out_size, void* d_ws, size_t ws_size,
                              hipStream_t stream) {
    const float* x    = (const float*)d_in[0];
    const float* h0   = (const float*)d_in[1];
    const float* c0   = (const float*)d_in[2];
    const float* Ux   = (const float*)d_in[3];
    const float* Wx   = (const float*)d_in[4];
    const float* Uh   = (const float*)d_in[5];
    const float* Vh   = (const float*)d_in[6];
    const float* bx   = (const float*)d_in[7];
    const float* bh   = (const float*)d_in[8];
    const float* diax = (const float*)d_in[9];
    const float* diah = (const float*)d_in[10];
    float* out = (float*)d_out;

    // Workspace (floats): P | hbuf | cbuf | tmp[3] | dvec_x | dvec_h  (~5.7 MB)
    float* w    = (float*)d_ws;
    float* P    = w;
    float* hbuf = P    + (size_t)M_ROWS * NWR;
    float* cbuf = hbuf + NB * NH;
    float* tmpb = cbuf + NB * NH;          // 3 rotating buffers of TMPSZ
    float* dvx  = tmpb + 3 * TMPSZ;
    float* dvh  = dvx  + 4 * NH;

    vml_init<<<dim3((NB * NH + 255) / 256), dim3(256), 0, stream>>>(h0, c0, hbuf, cbuf, tmpb);
    vml_dvec<<<dim3((4 * NH + 255) / 256), dim3(256), 0, stream>>>(Ux, Wx, Uh, Vh, dvx, dvh);
    // 1280 waves, 8 waves/block
    vml_xu_gemm<<<dim3(160), dim3(256), 0, stream>>>(x, Ux, P);
    // seed tmp buffer 0 from h0 (once)
    vml_tmp<<<dim3(12), dim3(64), 0, stream>>>(hbuf, Uh, tmpb);

    for (int t = 0; t < T_STEPS; ++t) {
        float* tcur  = tmpb + (size_t)(t % 3) * TMPSZ;
        float* tnext = tmpb + (size_t)((t + 1) % 3) * TMPSZ;
        float* tzero = tmpb + (size_t)((t + 2) % 3) * TMPSZ;
        // 192 waves, 2 waves/block -> spread across 96 WGPs; single launch per step
        vml_step<<<dim3(96), dim3(64), 0, stream>>>(t, P, Wx, Vh, Uh,
                                                    tcur, tnext, tzero, x,
                                                    hbuf, cbuf, dvx, dvh,
                                                    bx, bh, diax, diah, out);
    }
    vml_fin<<<dim3((NB * NH + 255) / 256), dim3(256), 0, stream>>>(hbuf, cbuf, out);
}